// GraphEncoder_43894565765354
// MI455X (gfx1250) — compile-verified
//
#include <hip/hip_runtime.h>

#define N_NODES 40000
#define N_EDGES 640000
#define D 128

typedef __attribute__((ext_vector_type(2))) float v2f;
typedef __attribute__((ext_vector_type(8))) float v8f;

// ---------------------------------------------------------------- utilities
__global__ void zero_f32_kernel(float* __restrict__ p, int n) {
  int i = blockIdx.x * blockDim.x + threadIdx.x;
  if (i < n) p[i] = 0.0f;
}

__global__ void degree_kernel(const int* __restrict__ dst,
                              float* __restrict__ deg, int ne) {
  int e = blockIdx.x * blockDim.x + threadIdx.x;
  if (e < ne) atomicAdd(&deg[dst[e]], 1.0f);
}

__global__ void invdeg_kernel(const float* __restrict__ deg,
                              float* __restrict__ invdeg, int n) {
  int i = blockIdx.x * blockDim.x + threadIdx.x;
  if (i < n) invdeg[i] = 1.0f / fmaxf(deg[i], 1.0f);
}

// One wave (32 lanes) per edge; each lane atomically adds a float4 (128 floats/row).
// All of feat + accum live in L2 (20.5 MB each), so these atomics resolve in L2.
__global__ void scatter_kernel(const float* __restrict__ feat,
                               const int* __restrict__ src,
                               const int* __restrict__ dst,
                               float* __restrict__ accum, int ne) {
  int tid = blockIdx.x * blockDim.x + threadIdx.x;
  int e = tid >> 5;
  int lane = tid & 31;
  if (e >= ne) return;
  int s = src[e];
  int d = dst[e];
  float4 v = ((const float4*)(feat + (size_t)s * D))[lane];
  float* o = accum + (size_t)d * D + lane * 4;
  atomicAdd(o + 0, v.x);
  atomicAdd(o + 1, v.y);
  atomicAdd(o + 2, v.z);
  atomicAdd(o + 3, v.w);
}

// Fused SAGE layer: Out = [relu]( (Accum*invdeg) @ Wl^T + bl + X @ Wr^T )
// Block = 256 thr = 8 waves; block b owns M-tile rows [16b,16b+16); wave w owns
// N-tile cols [16w,16w+16). One V_WMMA_F32_16X16X4_F32 per K-step of 4.
__global__ __launch_bounds__(256) void sage_gemm_kernel(
    const float* __restrict__ Accum, const float* __restrict__ invdeg,
    const float* __restrict__ X, const float* __restrict__ Wl,
    const float* __restrict__ bl, const float* __restrict__ Wr,
    float* __restrict__ Out, int apply_relu) {
  const int wave = threadIdx.x >> 5;
  const int lane = threadIdx.x & 31;
  const int lh = lane >> 4;   // 0: K=0,1 half / rows M=r ; 1: K=2,3 half / rows M=r+8
  const int ll = lane & 15;   // A-row / B-col / D-col within tile
  const int m0 = blockIdx.x * 16;
  const int n0 = wave * 16;

  const float* am = Accum + (size_t)(m0 + ll) * D;  // A operand rows (mean branch)
  const float* ax = X + (size_t)(m0 + ll) * D;      // A operand rows (root branch)
  const float* bL = Wl + (size_t)(n0 + ll) * D;     // B[k][n] = W[n][k] (contig in k)
  const float* bR = Wr + (size_t)(n0 + ll) * D;
  const float idg = invdeg[m0 + ll];                // fold mean-division into A load

  v8f c = {0.f, 0.f, 0.f, 0.f, 0.f, 0.f, 0.f, 0.f};

#pragma unroll
  for (int k = 0; k < D; k += 4) {
    const int ka = k + 2 * lh;
    v2f a = *(const v2f*)(am + ka);
    a = a * idg;
    v2f b = *(const v2f*)(bL + ka);
    c = __builtin_amdgcn_wmma_f32_16x16x4_f32(false, a, false, b, (short)0, c,
                                              false, false);
  }
#pragma unroll
  for (int k = 0; k < D; k += 4) {
    const int ka = k + 2 * lh;
    v2f a = *(const v2f*)(ax + ka);
    v2f b = *(const v2f*)(bR + ka);
    c = __builtin_amdgcn_wmma_f32_16x16x4_f32(false, a, false, b, (short)0, c,
                                              false, false);
  }

  const float bias = bl[n0 + ll];
#pragma unroll
  for (int r = 0; r < 8; ++r) {
    const int m = m0 + r + 8 * lh;
    float v = c[r] + bias;
    if (apply_relu) v = fmaxf(v, 0.0f);
    Out[(size_t)m * D + n0 + ll] = v;
  }
}

// ---------------------------------------------------------------- launcher
extern "C" void kernel_launch(void* const* d_in, const int* in_sizes, int n_in,
                              void* d_out, int out_size, void* d_ws,
                              size_t ws_size, hipStream_t stream) {
  const float* x = (const float*)d_in[0];
  const int* ei = (const int*)d_in[1];  // [2, N_EDGES]: row 0 = src, row 1 = dst
  const int* src = ei;
  const int* dst = ei + N_EDGES;
  const float* W1l = (const float*)d_in[2];
  const float* b1l = (const float*)d_in[3];
  const float* W1r = (const float*)d_in[4];
  const float* W2l = (const float*)d_in[5];
  const float* b2l = (const float*)d_in[6];
  const float* W2r = (const float*)d_in[7];
  float* out = (float*)d_out;

  char* ws = (char*)d_ws;
  float* accum = (float*)ws;                                    // N*D floats
  float* h = (float*)(ws + (size_t)N_NODES * D * 4);            // N*D floats
  float* deg = (float*)(ws + (size_t)N_NODES * D * 8);          // N floats
  float* invdeg = deg + N_NODES;                                // N floats

  const int ND = N_NODES * D;
  const int zb = (ND + 255) / 256;           // 20000 blocks
  const int eb = (N_EDGES + 255) / 256;      // 2500 blocks
  const int nb = (N_NODES + 255) / 256;      // 157 blocks
  const int sb = (N_EDGES * 32 + 255) / 256; // 80000 blocks (wave/edge)
  const int gb = N_NODES / 16;               // 2500 blocks (exact)

  // Degree + inverse degree (shared by both layers)
  zero_f32_kernel<<<nb, 256, 0, stream>>>(deg, N_NODES);
  degree_kernel<<<eb, 256, 0, stream>>>(dst, deg, N_EDGES);
  invdeg_kernel<<<nb, 256, 0, stream>>>(deg, invdeg, N_NODES);

  // Layer 1: aggregate x, fused GEMM + bias + ReLU -> h
  zero_f32_kernel<<<zb, 256, 0, stream>>>(accum, ND);
  scatter_kernel<<<sb, 256, 0, stream>>>(x, src, dst, accum, N_EDGES);
  sage_gemm_kernel<<<gb, 256, 0, stream>>>(accum, invdeg, x, W1l, b1l, W1r, h,
                                           1);

  // Layer 2: aggregate h, fused GEMM + bias -> out
  zero_f32_kernel<<<zb, 256, 0, stream>>>(accum, ND);
  scatter_kernel<<<sb, 256, 0, stream>>>(h, src, dst, accum, N_EDGES);
  sage_gemm_kernel<<<gb, 256, 0, stream>>>(accum, invdeg, h, W2l, b2l, W2r, out,
                                           0);
}